// HardMaskedAttention_51625506898051
// MI455X (gfx1250) — compile-verified
//
#include <hip/hip_runtime.h>
#include <hip/hip_bf16.h>

// HardMaskedAttention collapses: one-hot mask => softmax == 1.0 at the routed
// row. out = (gather(features, idx) @ Wv^T + bv) @ Wo^T + bo.
// Two 2048x1024x1024 fp32 GEMMs done with V_WMMA_F32_16X16X4_F32 (full fp32
// precision, matches the fp32 reference up to FMA ordering).
//
// Wave tile: 32 (M) x 64 (N) -> 8 accumulators, 2 A-fragments, 4 B-fragments
// per k-step: 6 b64 loads feed 8 WMMAs (B reused across both M-halves).

typedef float v2f __attribute__((ext_vector_type(2)));
typedef float v8f __attribute__((ext_vector_type(8)));

#define BB   4
#define KQ   512
#define VN   4096
#define CD   1024
#define MTOT (BB * KQ)          // 2048 rows
// grid: (MTOT/32) * (CD/64) = 64*16 = 1024 waves = 256 blocks of 128 threads

__global__ __launch_bounds__(128) void fused_linear_wmma_f32(
    const float* __restrict__ Abase,   // features (gather) or Y (no gather)
    const int*   __restrict__ ridx,    // routing indices (gather mode only)
    const float* __restrict__ W,       // (Cout=1024, Cin=1024) row-major
    const float* __restrict__ bias,    // (1024)
    float*       __restrict__ Out,     // (MTOT, 1024) row-major
    int gather)
{
    const int wid  = blockIdx.x * 4 + (threadIdx.x >> 5);  // 0..1023
    const int lane = threadIdx.x & 31;
    const int lm   = lane & 15;          // M (for A) / N (for B) within tile
    const int lk   = (lane >> 4) << 1;   // k sub-offset: 0 or 2

    const int mtile = wid >> 4;          // 0..63 (32-row M tiles)
    const int n0    = (wid & 15) << 6;   // 0,64,...,960
    const int m0    = mtile * 32;

    // ---- A row pointers (gather fused for GEMM-1) ----
    const int row0 = m0 + lm;            // rows m0..m0+15
    const int row1 = m0 + 16 + lm;       // rows m0+16..m0+31
    const float* __restrict__ arow0;
    const float* __restrict__ arow1;
    if (gather) {
        int b0i = row0 >> 9, q0 = row0 & 511;
        int b1i = row1 >> 9, q1 = row1 & 511;
        int v0 = ridx[b0i * KQ + q0];
        int v1 = ridx[b1i * KQ + q1];
        arow0 = Abase + ((size_t)b0i * VN + (size_t)v0) * CD;
        arow1 = Abase + ((size_t)b1i * VN + (size_t)v1) * CD;
    } else {
        arow0 = Abase + (size_t)row0 * CD;
        arow1 = Abase + (size_t)row1 * CD;
    }

    // ---- B row pointers: B[k][n] = W[n][k] (x @ W^T) ----
    const float* __restrict__ w0 = W + (size_t)(n0 +  0 + lm) * CD;
    const float* __restrict__ w1 = W + (size_t)(n0 + 16 + lm) * CD;
    const float* __restrict__ w2 = W + (size_t)(n0 + 32 + lm) * CD;
    const float* __restrict__ w3 = W + (size_t)(n0 + 48 + lm) * CD;

    v8f a00 = {}; v8f a01 = {}; v8f a02 = {}; v8f a03 = {};
    v8f a10 = {}; v8f a11 = {}; v8f a12 = {}; v8f a13 = {};

    for (int k = 0; k < CD; k += 4) {
        v2f a0 = *(const v2f*)(arow0 + k + lk);
        v2f a1 = *(const v2f*)(arow1 + k + lk);
        v2f b0 = *(const v2f*)(w0    + k + lk);
        v2f b1 = *(const v2f*)(w1    + k + lk);
        v2f b2 = *(const v2f*)(w2    + k + lk);
        v2f b3 = *(const v2f*)(w3    + k + lk);
        a00 = __builtin_amdgcn_wmma_f32_16x16x4_f32(false, a0, false, b0,
                                                    (short)0, a00, false, false);
        a01 = __builtin_amdgcn_wmma_f32_16x16x4_f32(false, a0, false, b1,
                                                    (short)0, a01, false, false);
        a02 = __builtin_amdgcn_wmma_f32_16x16x4_f32(false, a0, false, b2,
                                                    (short)0, a02, false, false);
        a03 = __builtin_amdgcn_wmma_f32_16x16x4_f32(false, a0, false, b3,
                                                    (short)0, a03, false, false);
        a10 = __builtin_amdgcn_wmma_f32_16x16x4_f32(false, a1, false, b0,
                                                    (short)0, a10, false, false);
        a11 = __builtin_amdgcn_wmma_f32_16x16x4_f32(false, a1, false, b1,
                                                    (short)0, a11, false, false);
        a12 = __builtin_amdgcn_wmma_f32_16x16x4_f32(false, a1, false, b2,
                                                    (short)0, a12, false, false);
        a13 = __builtin_amdgcn_wmma_f32_16x16x4_f32(false, a1, false, b3,
                                                    (short)0, a13, false, false);
    }

    // ---- epilogue: bias + store ----
    // C/D layout: N = lane%16, M = vgpr_i + 8*(lane/16)
    const float bz0 = bias[n0 +  0 + lm];
    const float bz1 = bias[n0 + 16 + lm];
    const float bz2 = bias[n0 + 32 + lm];
    const float bz3 = bias[n0 + 48 + lm];

    const int srow0 = m0 + ((lane >> 4) << 3);        // M base, first half
    float* __restrict__ o0 = Out + (size_t)srow0 * CD + n0 + lm;
    float* __restrict__ o1 = o0 + (size_t)16 * CD;    // second M half
    #pragma unroll
    for (int i = 0; i < 8; ++i) {
        o0[(size_t)i * CD +  0] = a00[i] + bz0;
        o0[(size_t)i * CD + 16] = a01[i] + bz1;
        o0[(size_t)i * CD + 32] = a02[i] + bz2;
        o0[(size_t)i * CD + 48] = a03[i] + bz3;
        o1[(size_t)i * CD +  0] = a10[i] + bz0;
        o1[(size_t)i * CD + 16] = a11[i] + bz1;
        o1[(size_t)i * CD + 32] = a12[i] + bz2;
        o1[(size_t)i * CD + 48] = a13[i] + bz3;
    }
}

extern "C" void kernel_launch(void* const* d_in, const int* in_sizes, int n_in,
                              void* d_out, int out_size, void* d_ws, size_t ws_size,
                              hipStream_t stream) {
    // setup_inputs order:
    // 0 query, 1 features, 2 routing_idx, 3 Wq, 4 bq, 5 Wk, 6 bk,
    // 7 Wv, 8 bv, 9 Wo, 10 bo
    const float* features = (const float*)d_in[1];
    const int*   ridx     = (const int*)  d_in[2];
    const float* Wv       = (const float*)d_in[7];
    const float* bv       = (const float*)d_in[8];
    const float* Wo       = (const float*)d_in[9];
    const float* bo       = (const float*)d_in[10];
    float* out = (float*)d_out;
    float* Y   = (float*)d_ws;   // 2048*1024*4 = 8 MB intermediate

    dim3 grid(256);   // 1024 waves / 4 waves per block
    dim3 block(128);  // 4 wave32

    // GEMM 1: Y = gather(features, ridx) @ Wv^T + bv
    fused_linear_wmma_f32<<<grid, block, 0, stream>>>(features, ridx, Wv, bv, Y, 1);
    // GEMM 2: out = Y @ Wo^T + bo
    fused_linear_wmma_f32<<<grid, block, 0, stream>>>(Y, ridx, Wo, bo, out, 0);
}